// TemporalSegmentAttention_70471823393384
// MI455X (gfx1250) — compile-verified
//
#include <hip/hip_runtime.h>
#include <hip/hip_bf16.h>

typedef __attribute__((ext_vector_type(2))) float v2f;
typedef __attribute__((ext_vector_type(8))) float v8f;

#define B_   8
#define T_   4096
#define BC_  256
#define RC_  64
#define TB_  7
#define NB0_ 4

// ---------------------------------------------------------------------------
// Zero the gap accumulator (d_ws is poisoned with 0xAA by the harness).
// ---------------------------------------------------------------------------
__global__ void init_kernel(float* __restrict__ gap) {
    int i = threadIdx.x;
#pragma unroll
    for (int j = 0; j < 8; ++j) gap[i + j * 256] = 0.0f;  // 2048 floats
}

// ---------------------------------------------------------------------------
// gap[b][c] += (1/T) * sum_t x[b][ch][t], folded over blocks (ch & 255).
// One 256-thread block per (b, ch) row; float4 (B128) coalesced loads.
// ---------------------------------------------------------------------------
__global__ __launch_bounds__(256) void gap_kernel(const float* __restrict__ x,
                                                  float* __restrict__ gap,
                                                  int nch) {
    int row = blockIdx.x;
    int b   = row / nch;
    int ch  = row - b * nch;
    const float4* p = (const float4*)(x + (size_t)row * T_);

    float s = 0.0f;
#pragma unroll
    for (int i = 0; i < 4; ++i) {
        float4 v = p[threadIdx.x + i * 256];
        s += v.x + v.y + v.z + v.w;
    }
    // wave32 reduction
#pragma unroll
    for (int off = 16; off > 0; off >>= 1) s += __shfl_xor(s, off, 32);

    __shared__ float wsum[8];
    int lane = threadIdx.x & 31, wv = threadIdx.x >> 5;
    if (lane == 0) wsum[wv] = s;
    __syncthreads();
    if (wv == 0) {
        s = (lane < 8) ? wsum[lane] : 0.0f;
#pragma unroll
        for (int off = 4; off > 0; off >>= 1) s += __shfl_xor(s, off, 32);
        if (lane == 0)
            atomicAdd(&gap[b * BC_ + (ch & (BC_ - 1))], s * (1.0f / (float)T_));
    }
}

// ---------------------------------------------------------------------------
// Single-block attention head: two fp32 WMMA GEMMs + softmax over blocks.
//   h[8,64]      = relu((gap[8,256] @ W1[256,64] + b1) * gamma/sqrt(1+eps) + beta)
//   sc[7,8,256]  = h @ Wh[k] + bh[k]
//   att[8,7,256] = softmax_k(sc)
// 256 threads = 8 wave32 waves; M=8 padded to 16 with zero rows.
// gap_s (phase 1 only) aliases sc_s (phase 2+) to stay at ~60 KB LDS.
// ---------------------------------------------------------------------------
__global__ __launch_bounds__(256) void atten_kernel(
    const float* __restrict__ gap, const float* __restrict__ W1,
    const float* __restrict__ b1, const float* __restrict__ gamma,
    const float* __restrict__ beta, const float* __restrict__ Wh,
    const float* __restrict__ bh, float* __restrict__ att) {
    __shared__ union {
        float gap[16 * BC_];          // 16 KB, zero-padded rows 8..15 (phase 1)
        float sc[TB_ * B_ * BC_];     // 56 KB scores (phase 2+)
    } sm;
    __shared__ float h_s[16][RC_];    // 4 KB (rows 8..15 forced zero)

    const int tid  = threadIdx.x;
    const int lane = tid & 31;
    const int wv   = tid >> 5;
    const int m    = lane & 15;       // matrix row (A) / column (B,C,D)
    const int half = lane >> 4;       // 0: K/M low half, 1: high half

    // ---- Stage zero-padded gap[16][256] into LDS -------------------------
    for (int i = tid; i < 16 * BC_; i += 256)
        sm.gap[i] = ((i >> 8) < B_) ? gap[i] : 0.0f;
    __syncthreads();

    // ---- Phase 1: h = relu(bn(gap @ W1 + b1)); waves 0..3, N-tile = wv ----
    if (wv < 4) {
        v8f acc = {0.f, 0.f, 0.f, 0.f, 0.f, 0.f, 0.f, 0.f};
        const int n = wv * 16 + m;
        for (int k0 = 0; k0 < BC_; k0 += 4) {
            const int kk = k0 + half * 2;
            const v2f a = *(const v2f*)&sm.gap[m * BC_ + kk];  // A: 16x4 f32
            v2f bf;
            bf.x = W1[kk * RC_ + n];                           // B: 4x16 f32
            bf.y = W1[(kk + 1) * RC_ + n];
            acc = __builtin_amdgcn_wmma_f32_16x16x4_f32(
                false, a, false, bf, (short)0, acc, false, false);
        }
        const int n2  = wv * 16 + m;
        const float scl = gamma[n2] * rsqrtf(1.0f + 1e-5f);
        const float bta = beta[n2];
        const float bb  = b1[n2];
        if (half == 0) {              // C/D rows 0..7 live in lanes 0..15
#pragma unroll
            for (int r = 0; r < 8; ++r)
                h_s[r][n2] = fmaxf((acc[r] + bb) * scl + bta, 0.0f);
        } else {                      // rows 8..15: zero padding
#pragma unroll
            for (int r = 0; r < 8; ++r) h_s[r + 8][n2] = 0.0f;
        }
    }
    __syncthreads();                  // gap_s dead; sc_s live from here

    // ---- Phase 2: scores[k][b][c] = h @ Wh[k] + bh[k]; 112 tiles / 8 waves
    for (int t = wv; t < TB_ * 16; t += 8) {                  // uniform trips
        const int k  = t >> 4;
        const int nt = t & 15;
        const int n  = nt * 16 + m;
        v8f acc = {0.f, 0.f, 0.f, 0.f, 0.f, 0.f, 0.f, 0.f};
        for (int k0 = 0; k0 < RC_; k0 += 4) {
            const int kk = k0 + half * 2;
            const v2f a = *(const v2f*)&h_s[m][kk];
            v2f bf;
            bf.x = Wh[(k * RC_ + kk) * BC_ + n];
            bf.y = Wh[(k * RC_ + kk + 1) * BC_ + n];
            acc = __builtin_amdgcn_wmma_f32_16x16x4_f32(
                false, a, false, bf, (short)0, acc, false, false);
        }
        if (half == 0) {              // rows 0..7 == batch entries
            const float bias = bh[k * BC_ + n];
#pragma unroll
            for (int r = 0; r < 8; ++r)
                sm.sc[(k * B_ + r) * BC_ + n] = acc[r] + bias;
        }
    }
    __syncthreads();

    // ---- Phase 3: softmax over k, write att[b][k][c] (LOWEST==0) ---------
    for (int idx = tid; idx < B_ * BC_; idx += 256) {
        const int b = idx >> 8;
        const int c = idx & (BC_ - 1);
        float e[TB_], mx = -3.402823466e+38f;
#pragma unroll
        for (int k = 0; k < TB_; ++k) {
            e[k] = sm.sc[(k * B_ + b) * BC_ + c];
            mx = fmaxf(mx, e[k]);
        }
        float sum = 0.0f;
#pragma unroll
        for (int k = 0; k < TB_; ++k) { e[k] = __expf(e[k] - mx); sum += e[k]; }
        const float inv = 1.0f / sum;
#pragma unroll
        for (int k = 0; k < TB_; ++k)
            att[(b * TB_ + k) * BC_ + c] = e[k] * inv;
    }
}

// ---------------------------------------------------------------------------
// out[b][ch][t] = x[b][ch][t] * att[b][k_off + ch/256][ch%256]
// One block per row; block-uniform attention scalar; float4 streaming.
// ---------------------------------------------------------------------------
__global__ __launch_bounds__(256) void scale_kernel(const float* __restrict__ x,
                                                    float* __restrict__ out,
                                                    const float* __restrict__ att,
                                                    int nch, int k_off) {
    int row = blockIdx.x;
    int b   = row / nch;
    int ch  = row - b * nch;
    const float a =
        att[(b * TB_ + k_off + (ch >> 8)) * BC_ + (ch & (BC_ - 1))];
    const float4* p = (const float4*)(x + (size_t)row * T_);
    float4* q = (float4*)(out + (size_t)row * T_);
#pragma unroll
    for (int i = 0; i < 4; ++i) {
        float4 v = p[threadIdx.x + i * 256];
        v.x *= a; v.y *= a; v.z *= a; v.w *= a;
        q[threadIdx.x + i * 256] = v;
    }
}

extern "C" void kernel_launch(void* const* d_in, const int* in_sizes, int n_in,
                              void* d_out, int out_size, void* d_ws, size_t ws_size,
                              hipStream_t stream) {
    const float* x0    = (const float*)d_in[0];  // [8,1024,4096]
    const float* x1    = (const float*)d_in[1];  // [8, 768,4096]
    const float* W1    = (const float*)d_in[2];  // [256,64]
    const float* b1    = (const float*)d_in[3];  // [64]
    const float* gamma = (const float*)d_in[4];  // [64]
    const float* beta  = (const float*)d_in[5];  // [64]
    const float* Wh    = (const float*)d_in[6];  // [7,64,256]
    const float* bh    = (const float*)d_in[7];  // [7,256]

    float* gap = (float*)d_ws;        // 2048 floats
    float* att = gap + B_ * BC_;      // 8*7*256 = 14336 floats

    float* out0 = (float*)d_out;                                   // 8*1024*4096
    float* out1 = out0 + (size_t)B_ * 1024 * T_;                   // 8* 768*4096

    init_kernel<<<1, 256, 0, stream>>>(gap);
    gap_kernel<<<B_ * 1024, 256, 0, stream>>>(x0, gap, 1024);
    gap_kernel<<<B_ * 768, 256, 0, stream>>>(x1, gap, 768);
    atten_kernel<<<1, 256, 0, stream>>>(gap, W1, b1, gamma, beta, Wh, bh, att);
    scale_kernel<<<B_ * 1024, 256, 0, stream>>>(x0, out0, att, 1024, 0);
    scale_kernel<<<B_ * 768, 256, 0, stream>>>(x1, out1, att, 768, NB0_);
}